// DyRepNode_month_83107617178205
// MI455X (gfx1250) — compile-verified
//
#include <hip/hip_runtime.h>
#include <math.h>

#define N_NODES 50000
#define HID     128
#define BATCH   256
#define RS      132            // padded LDS row stride in floats (132 % 64 = 4 -> conflict-free frag reads)
#define NTILES  (N_NODES / 16) // 3125

// workspace layout (float offsets)
#define WS_PROB   0
#define WS_PACK   51200                 // 8B-aligned, leaves room for prob[N]
#define PK_SZ     (8 * 32 * 64)         // 16384 floats per packed 128x128 weight

typedef float v2f __attribute__((ext_vector_type(2)));
typedef float v8f __attribute__((ext_vector_type(8)));

__device__ __forceinline__ float lrelu_f(float x) { return x > 0.f ? x : 0.01f * x; }

// Repack W[128][128] (row-major, fan_in x fan_out) into WMMA B-fragment order:
// fragment (j,k) = 64 floats, lane L holds {W[4k+2*hi][16j+l15], W[4k+2*hi+1][16j+l15]}
// stored contiguously -> one global_load_b64 per WMMA, 256B/wave coalesced.
__global__ __launch_bounds__(256) void pack_kernel(const float* __restrict__ W,
                                                   float* __restrict__ Wp)
{
  const int t    = blockIdx.x * 256 + threadIdx.x;   // 0..8191
  const int lane = t & 31;
  const int k    = (t >> 5) & 31;
  const int j    = (t >> 10) & 7;
  const int hi   = lane >> 4, l15 = lane & 15;
  const int col  = j * 16 + l15;
  const int kr   = k * 4 + hi * 2;
  float2 v;
  v.x = W[(size_t)kr * HID + col];
  v.y = W[(size_t)(kr + 1) * HID + col];
  *(float2*)(Wp + 2 * (size_t)t) = v;
}

// One wave: dst[16 x 128] = lrelu(src[16 x 128] @ W + b), W pre-packed.
// A fragments for all 32 K-steps are held in registers (64 VGPRs) and reused
// across the 8 column tiles; one b64 weight load per WMMA.
__device__ __forceinline__ void layer_16xH(const float* __restrict__ Wp,
                                           const float* __restrict__ Bv,
                                           const float* src, float* dst,
                                           int lane)
{
  const int l15 = lane & 15;
  const int hi  = lane >> 4;

  // preload A: lane m holds {A[m, 4k+2hi], A[m, 4k+2hi+1]} per k
  v2f a[32];
  const float* arow = src + l15 * RS + hi * 2;
#pragma unroll
  for (int k = 0; k < 32; ++k) a[k] = *(const v2f*)(arow + 4 * k);

  const float* wl = Wp + 2 * lane;                  // lane's slot inside each 64-float fragment
  for (int j = 0; j < 8; ++j) {
    const float bv = Bv[j * 16 + l15];
    v8f acc = {bv, bv, bv, bv, bv, bv, bv, bv};
    const float* wj = wl + (size_t)j * 32 * 64;
#pragma unroll
    for (int k = 0; k < 32; ++k) {
      v2f b = *(const v2f*)(wj + k * 64);           // global_load_b64, contiguous 256B/wave
      acc = __builtin_amdgcn_wmma_f32_16x16x4_f32(false, a[k], false, b,
                                                  (short)0, acc, false, false);
    }
    // D layout: vgpr r, lanes 0-15 -> row r; lanes 16-31 -> row 8+r
    const int rbase = hi * 8;
    float* drow = dst + j * 16 + l15;
#pragma unroll
    for (int r = 0; r < 8; ++r) {
      float x = acc[r];
      drow[(rbase + r) * RS] = lrelu_f(x);
    }
  }
}

// prob head: per 16-node tile, 4x(H->H, lrelu) then (H->1, sigmoid) -> prob[N]
__global__ __launch_bounds__(128) void prob_kernel(
    const float* __restrict__ z,
    const float* __restrict__ w1p, const float* __restrict__ b1,
    const float* __restrict__ w2p, const float* __restrict__ b2,
    const float* __restrict__ w3p, const float* __restrict__ b3,
    const float* __restrict__ w4p, const float* __restrict__ b4,
    const float* __restrict__ w5,  const float* __restrict__ b5,
    float* __restrict__ prob)
{
  __shared__ float lds[4][2][16 * RS];
  const int wave = threadIdx.x >> 5;
  const int lane = threadIdx.x & 31;

  int tile = blockIdx.x * 4 + wave;
  const bool valid = (tile < NTILES);
  if (!valid) tile = NTILES - 1;             // clamp tail waves: redundant compute, masked store
  const int row0 = tile * 16;

  float* bufA = &lds[wave][0][0];
  float* bufB = &lds[wave][1][0];

  // stage the 16x128 z tile into LDS (one b128 per lane per row)
#pragma unroll 4
  for (int r = 0; r < 16; ++r) {
    const float4 v = *(const float4*)(z + (size_t)(row0 + r) * HID + lane * 4);
    *(float4*)(bufA + r * RS + lane * 4) = v;
  }

  layer_16xH(w1p, b1, bufA, bufB, lane);
  layer_16xH(w2p, b2, bufB, bufA, lane);
  layer_16xH(w3p, b3, bufA, bufB, lane);
  layer_16xH(w4p, b4, bufB, bufA, lane);

  // H->1 head: lane (l15,hi) reduces half a row, combine halves via shfl_xor(16)
  const int l15 = lane & 15, hi = lane >> 4;
  const float* hrow = bufA + l15 * RS + hi * 64;
  const float* w5h  = w5 + hi * 64;
  float s = 0.f;
#pragma unroll 8
  for (int c = 0; c < 64; ++c) s = fmaf(hrow[c], w5h[c], s);
  s += __shfl_xor(s, 16, 32);
  if (valid && lane < 16) {
    const float x = s + b5[0];
    prob[row0 + l15] = 1.f / (1.f + expf(-x));
  }
}

// time head: gather z[u] (256 rows), 2x(H->H, lrelu) then (H->1 linear) -> out[B]
__global__ __launch_bounds__(128) void time_kernel(
    const float* __restrict__ z, const int* __restrict__ u,
    const float* __restrict__ w1p, const float* __restrict__ b1,
    const float* __restrict__ w2p, const float* __restrict__ b2,
    const float* __restrict__ w3,  const float* __restrict__ b3,
    float* __restrict__ tout)
{
  __shared__ float lds[4][2][16 * RS];
  const int wave = threadIdx.x >> 5;
  const int lane = threadIdx.x & 31;
  const int tile = blockIdx.x * 4 + wave;    // 16 tiles exactly (grid = 4 blocks)
  const int row0 = tile * 16;

  float* bufA = &lds[wave][0][0];
  float* bufB = &lds[wave][1][0];

#pragma unroll 4
  for (int r = 0; r < 16; ++r) {
    const int node = u[row0 + r];            // scalar broadcast load
    const float4 v = *(const float4*)(z + (size_t)node * HID + lane * 4);
    *(float4*)(bufA + r * RS + lane * 4) = v;
  }

  layer_16xH(w1p, b1, bufA, bufB, lane);
  layer_16xH(w2p, b2, bufB, bufA, lane);

  const int l15 = lane & 15, hi = lane >> 4;
  const float* hrow = bufA + l15 * RS + hi * 64;
  const float* w3h  = w3 + hi * 64;
  float s = 0.f;
#pragma unroll 8
  for (int c = 0; c < 64; ++c) s = fmaf(hrow[c], w3h[c], s);
  s += __shfl_xor(s, 16, 32);
  if (lane < 16) tout[row0 + l15] = s + b3[0];
}

// broadcast prob[N] -> out[B, N]; pure HBM-bandwidth kernel, fully coalesced b128 stores
__global__ __launch_bounds__(256) void bcast_kernel(const float* __restrict__ prob,
                                                    float* __restrict__ out)
{
  const int col = blockIdx.x * 1024 + threadIdx.x * 4;
  if (col + 4 <= N_NODES) {
    const float4 v = *(const float4*)(prob + col);
    *(float4*)(out + (size_t)blockIdx.y * N_NODES + col) = v;
  }
}

extern "C" void kernel_launch(void* const* d_in, const int* in_sizes, int n_in,
                              void* d_out, int out_size, void* d_ws, size_t ws_size,
                              hipStream_t stream) {
  const float* z   = (const float*)d_in[0];
  const int*   u   = (const int*)  d_in[1];
  const float* pw1 = (const float*)d_in[2];  const float* pb1 = (const float*)d_in[3];
  const float* pw2 = (const float*)d_in[4];  const float* pb2 = (const float*)d_in[5];
  const float* pw3 = (const float*)d_in[6];  const float* pb3 = (const float*)d_in[7];
  const float* pw4 = (const float*)d_in[8];  const float* pb4 = (const float*)d_in[9];
  const float* pw5 = (const float*)d_in[10]; const float* pb5 = (const float*)d_in[11];
  const float* tw1 = (const float*)d_in[12]; const float* tb1 = (const float*)d_in[13];
  const float* tw2 = (const float*)d_in[14]; const float* tb2 = (const float*)d_in[15];
  const float* tw3 = (const float*)d_in[16]; const float* tb3 = (const float*)d_in[17];

  float* out  = (float*)d_out;               // [BATCH*N_NODES] prob_list ++ [BATCH] time_list
  float* ws   = (float*)d_ws;
  float* prob = ws + WS_PROB;                // N_NODES floats
  float* w1p  = ws + WS_PACK + 0 * PK_SZ;
  float* w2p  = ws + WS_PACK + 1 * PK_SZ;
  float* w3p  = ws + WS_PACK + 2 * PK_SZ;
  float* w4p  = ws + WS_PACK + 3 * PK_SZ;
  float* t1p  = ws + WS_PACK + 4 * PK_SZ;
  float* t2p  = ws + WS_PACK + 5 * PK_SZ;

  // 0) repack the six 128x128 weight matrices into WMMA fragment order (deterministic, every call)
  pack_kernel<<<32, 256, 0, stream>>>(pw1, w1p);
  pack_kernel<<<32, 256, 0, stream>>>(pw2, w2p);
  pack_kernel<<<32, 256, 0, stream>>>(pw3, w3p);
  pack_kernel<<<32, 256, 0, stream>>>(pw4, w4p);
  pack_kernel<<<32, 256, 0, stream>>>(tw1, t1p);
  pack_kernel<<<32, 256, 0, stream>>>(tw2, t2p);

  // 1) prob[N] via WMMA MLP (3125 tiles, 4 waves/block)
  prob_kernel<<<(NTILES + 3) / 4, 128, 0, stream>>>(
      z, w1p, pb1, w2p, pb2, w3p, pb3, w4p, pb4, pw5, pb5, prob);

  // 2) time_list[B] via WMMA MLP on gathered rows (16 tiles)
  time_kernel<<<4, 128, 0, stream>>>(
      z, u, t1p, tb1, t2p, tb2, tw3, tb3, out + (size_t)BATCH * N_NODES);

  // 3) broadcast prob -> prob_list [B, N]
  dim3 bgrid((N_NODES + 1023) / 1024, BATCH);
  bcast_kernel<<<bgrid, 256, 0, stream>>>(prob, out);
}